// GraphConvolutionalNetwork_41437844471994
// MI455X (gfx1250) — compile-verified
//
#include <hip/hip_runtime.h>

typedef __attribute__((ext_vector_type(2))) float f32x2;
typedef __attribute__((ext_vector_type(8))) float f32x8;

#define N_FEAT 128

// ---------------------------------------------------------------------------
// GEMM: Y[N x 128] = X[N x 128] * W[128 x 128] using V_WMMA_F32_16X16X4_F32.
// grid.x = N/16 row stripes, 256 threads = 8 waves, wave w -> column tile w.
// A 16x4 f32 layout: lane l: m = l&15, VGPR r holds K = 2*(l>>4) + r.
// B 4x16 f32 layout: lane l: n = l&15, VGPR r holds K = 2*(l>>4) + r.
// C/D 16x16 f32:     lane l: n = l&15, VGPR v holds row v + 8*(l>>4).
// ---------------------------------------------------------------------------
__global__ __launch_bounds__(256) void gcn_gemm_wmma_f32(
    const float* __restrict__ X, const float* __restrict__ W,
    float* __restrict__ Y) {
  const int lane = threadIdx.x & 31;
  const int wv   = threadIdx.x >> 5;   // 0..7 -> 16-col tile
  const int m0   = blockIdx.x << 4;    // 16-row stripe
  const int n0   = wv << 4;
  const int r    = lane & 15;
  const int hh   = lane >> 4;          // 0 or 1

  const float* xrow = X + (size_t)(m0 + r) * N_FEAT;
  f32x8 acc = {0.f, 0.f, 0.f, 0.f, 0.f, 0.f, 0.f, 0.f};

#pragma unroll 8
  for (int k0 = 0; k0 < N_FEAT; k0 += 4) {
    const int ka = k0 + 2 * hh;
    f32x2 a = *(const f32x2*)(xrow + ka);
    f32x2 b;
    b.x = W[(size_t)(ka + 0) * N_FEAT + n0 + r];
    b.y = W[(size_t)(ka + 1) * N_FEAT + n0 + r];
    acc = __builtin_amdgcn_wmma_f32_16x16x4_f32(false, a, false, b, (short)0,
                                                acc, false, false);
  }

  float* yout = Y + (size_t)(m0 + 8 * hh) * N_FEAT + n0 + r;
#pragma unroll
  for (int v = 0; v < 8; ++v) yout[(size_t)v * N_FEAT] = acc[v];
}

// ---------------------------------------------------------------------------
// Degree / normalization kernels
// ---------------------------------------------------------------------------
__global__ void gcn_deg_init(float* __restrict__ deg, int n) {
  int i = blockIdx.x * blockDim.x + threadIdx.x;
  if (i < n) deg[i] = 1.0f;  // self-loop contribution
}

__global__ void gcn_deg_accum(const int* __restrict__ dst,
                              float* __restrict__ deg, int E) {
  int i = blockIdx.x * blockDim.x + threadIdx.x;
  int stride = gridDim.x * blockDim.x;
  for (int e = i; e < E; e += stride) atomicAdd(&deg[dst[e]], 1.0f);
}

__global__ void gcn_dinv(float* __restrict__ deg, int n) {
  int i = blockIdx.x * blockDim.x + threadIdx.x;
  if (i < n) {
    float d = deg[i];
    deg[i] = (d > 0.0f) ? rsqrtf(d) : 0.0f;  // dinv in place
  }
}

// ---------------------------------------------------------------------------
// agg[i,:] = t[i,:] * dinv[i]^2   (self-loop term; also zero-initializes agg)
// ---------------------------------------------------------------------------
__global__ void gcn_self_init(const float* __restrict__ t,
                              const float* __restrict__ dinv,
                              float* __restrict__ agg, int n_elems) {
  int i = blockIdx.x * blockDim.x + threadIdx.x;
  if (i < n_elems) {
    float di = dinv[i >> 7];  // node = i / 128
    agg[i] = t[i] * di * di;
  }
}

// ---------------------------------------------------------------------------
// Edge scatter-gather: one wave32 per edge; lane l moves float4 of features
// 4l..4l+3.  agg[dst,:] += t[src,:] * dinv[src]*dinv[dst]  via f32 atomics.
// ---------------------------------------------------------------------------
__global__ __launch_bounds__(256) void gcn_edge_agg(
    const float* __restrict__ t, const int* __restrict__ src,
    const int* __restrict__ dst, const float* __restrict__ dinv,
    float* __restrict__ agg, int E) {
  const int lane = threadIdx.x & 31;
  int wave = (blockIdx.x * blockDim.x + threadIdx.x) >> 5;
  const int nwaves = (gridDim.x * blockDim.x) >> 5;

  for (int e = wave; e < E; e += nwaves) {
    int en = e + nwaves;
    if (en < E) {  // hint the next edge indices into cache
      __builtin_prefetch(&src[en], 0, 0);
      __builtin_prefetch(&dst[en], 0, 0);
    }
    int s = src[e];
    int d = dst[e];
    float norm = dinv[s] * dinv[d];
    const float4 v = *(const float4*)(t + (size_t)s * N_FEAT + lane * 4);
    float* o = agg + (size_t)d * N_FEAT + lane * 4;
    atomicAdd(o + 0, v.x * norm);
    atomicAdd(o + 1, v.y * norm);
    atomicAdd(o + 2, v.z * norm);
    atomicAdd(o + 3, v.w * norm);
  }
}

// ---------------------------------------------------------------------------
// Elementwise epilogues
// ---------------------------------------------------------------------------
__global__ void gcn_bias_relu(const float* __restrict__ agg,
                              const float* __restrict__ bias,
                              float* __restrict__ h, int n_elems) {
  int i = blockIdx.x * blockDim.x + threadIdx.x;
  if (i < n_elems) {
    float v = agg[i] + bias[i & (N_FEAT - 1)];
    h[i] = v > 0.0f ? v : 0.0f;
  }
}

__global__ void gcn_bias_add(float* __restrict__ out,
                             const float* __restrict__ bias, int n_elems) {
  int i = blockIdx.x * blockDim.x + threadIdx.x;
  if (i < n_elems) out[i] += bias[i & (N_FEAT - 1)];
}

// ---------------------------------------------------------------------------
// Launch: deg -> dinv -> [gemm, self, edges, bias(+relu)] x 2 layers
// ---------------------------------------------------------------------------
extern "C" void kernel_launch(void* const* d_in, const int* in_sizes, int n_in,
                              void* d_out, int out_size, void* d_ws,
                              size_t ws_size, hipStream_t stream) {
  const float* x  = (const float*)d_in[0];
  const int*   ei = (const int*)d_in[1];
  const float* W1 = (const float*)d_in[2];
  const float* b1 = (const float*)d_in[3];
  const float* W2 = (const float*)d_in[4];
  const float* b2 = (const float*)d_in[5];

  const int N = in_sizes[0] / N_FEAT;   // 100000 (divisible by 16)
  const int E = in_sizes[1] / 2;        // 1600000
  const int* src = ei;
  const int* dst = ei + E;
  const int NE = N * N_FEAT;

  // Workspace layout: dinv | t | agg
  char* ws = (char*)d_ws;
  size_t degBytes = (((size_t)N * sizeof(float)) + 255) & ~(size_t)255;
  float* dinv = (float*)ws;
  float* t    = (float*)(ws + degBytes);
  float* agg  = (float*)(ws + degBytes + (size_t)NE * sizeof(float));
  float* out  = (float*)d_out;

  const int TB = 256;
  dim3 nodeGrid((N + TB - 1) / TB);
  dim3 elemGrid((NE + TB - 1) / TB);
  dim3 gemmGrid(N / 16);
  dim3 edgeGrid(4096);

  // Degree + normalization (shared by both layers)
  gcn_deg_init<<<nodeGrid, TB, 0, stream>>>(dinv, N);
  gcn_deg_accum<<<512, TB, 0, stream>>>(dst, dinv, E);
  gcn_dinv<<<nodeGrid, TB, 0, stream>>>(dinv, N);

  // Layer 1: t = x@W1 ; agg = A_hat @ t ; t = relu(agg + b1)
  gcn_gemm_wmma_f32<<<gemmGrid, TB, 0, stream>>>(x, W1, t);
  gcn_self_init<<<elemGrid, TB, 0, stream>>>(t, dinv, agg, NE);
  gcn_edge_agg<<<edgeGrid, TB, 0, stream>>>(t, src, dst, dinv, agg, E);
  gcn_bias_relu<<<elemGrid, TB, 0, stream>>>(agg, b1, t, NE);

  // Layer 2: agg = t@W2 ; out = A_hat @ agg ; out += b2
  gcn_gemm_wmma_f32<<<gemmGrid, TB, 0, stream>>>(t, W2, agg);
  gcn_self_init<<<elemGrid, TB, 0, stream>>>(agg, dinv, out, NE);
  gcn_edge_agg<<<edgeGrid, TB, 0, stream>>>(agg, src, dst, dinv, out, E);
  gcn_bias_add<<<elemGrid, TB, 0, stream>>>(out, b2, NE);
}